// Force_55525337202860
// MI455X (gfx1250) — compile-verified
//
#include <hip/hip_runtime.h>
#include <hip/hip_bf16.h>
#include <math.h>

// ---------------------------------------------------------------------------
// Problem constants (match reference)
// ---------------------------------------------------------------------------
#define E_EDGES 1600000
#define FDIM    64
#define N_NODES 50000
#define BN_EPS  1e-5f

typedef float v2f __attribute__((ext_vector_type(2)));
typedef float v8f __attribute__((ext_vector_type(8)));

#ifndef __has_builtin
#define __has_builtin(x) 0
#endif

#if __has_builtin(__builtin_amdgcn_wmma_f32_16x16x4_f32)
#define HAVE_WMMA_F32 1
#else
#define HAVE_WMMA_F32 0
#endif

__device__ __forceinline__ float softplus_f(float x) {
    // numerically stable softplus: max(x,0) + log1p(exp(-|x|))
    return fmaxf(x, 0.f) + log1pf(expf(-fabsf(x)));
}

// ---------------------------------------------------------------------------
// init: zero force output and BN-stat accumulators
// ---------------------------------------------------------------------------
__global__ void init_zero_kernel(float* __restrict__ out, int n_out,
                                 float* __restrict__ stats, int n_stats) {
    int i = blockIdx.x * blockDim.x + threadIdx.x;
    if (i < n_out)   out[i]   = 0.f;
    if (i < n_stats) stats[i] = 0.f;
}

// ---------------------------------------------------------------------------
// Layer kernel: y = act(x) @ W + b, where act = BN(prev stats)+softplus
// (identity for layer 0). Accumulates per-feature sum / sumsq of y into
// statsOut (pre-BN stats consumed by the *next* kernel). In-place safe
// (x == y): each wave reads its 16 rows fully before storing them.
// ---------------------------------------------------------------------------
template <int APPLY_ACT>
__launch_bounds__(256)
__global__ void gemm_bn_layer(const float* x, float* y,
                              const float* __restrict__ Wg,        // [64][64] (k,n)
                              const float* __restrict__ bias,      // [64]
                              const float* __restrict__ gPrev,     // [64] or null
                              const float* __restrict__ bPrev,     // [64] or null
                              const float* __restrict__ statsPrev, // [128] or null
                              float* __restrict__ statsOut)        // [128] sum|sumsq
{
    __shared__ float ldsWt[FDIM * FDIM];   // W transposed: Wt[n][k]
    __shared__ float ldsBias[FDIM];
    __shared__ float preScale[FDIM];
    __shared__ float preShift[FDIM];
    __shared__ float ldsSum[FDIM];
    __shared__ float ldsSq[FDIM];

    const int tid = threadIdx.x;

    for (int i = tid; i < FDIM * FDIM; i += 256) {
        int k = i >> 6, n = i & 63;
        ldsWt[n * FDIM + k] = Wg[i];
    }
    if (tid < FDIM) {
        ldsBias[tid] = bias[tid];
        ldsSum[tid]  = 0.f;
        ldsSq[tid]   = 0.f;
        if (APPLY_ACT) {
            const float inv_e = 1.f / (float)E_EDGES;
            float mu   = statsPrev[tid] * inv_e;
            float var  = statsPrev[FDIM + tid] * inv_e - mu * mu;
            float rstd = rsqrtf(var + BN_EPS);
            float g    = gPrev[tid] * rstd;
            preScale[tid] = g;
            preShift[tid] = bPrev[tid] - mu * g;
        } else {
            preScale[tid] = 1.f;
            preShift[tid] = 0.f;
        }
    }
    __syncthreads();

    const int lane    = tid & 31;
    const int wave    = tid >> 5;
    const int waveGid = blockIdx.x * 8 + wave;
    const int nWaves  = gridDim.x * 8;
    const int mrow    = lane & 15;
    const int halfSel = lane >> 4;      // 0: K pair {0,1}; 1: K pair {2,3}
    const int kOff    = halfSel * 2;

    float sumAcc[4] = {0.f, 0.f, 0.f, 0.f};
    float sqAcc[4]  = {0.f, 0.f, 0.f, 0.f};

    const int    nTiles    = E_EDGES / 16;
    const size_t stepElems = (size_t)nWaves * 16 * FDIM;

    // one 64-bit base per stream; all inner accesses use immediate offsets
    const float* xr = x + (size_t)(waveGid * 16 + mrow) * FDIM + kOff;
    float*       yp = y + (size_t)(waveGid * 16 + halfSel * 8) * FDIM + mrow;

    for (int t = waveGid; t < nTiles; t += nWaves, xr += stepElems, yp += stepElems) {
        v8f acc0 = {}, acc1 = {}, acc2 = {}, acc3 = {};

#if HAVE_WMMA_F32
        #pragma unroll
        for (int kb = 0; kb < 16; ++kb) {
            const int k0 = kb * 4 + kOff;
            v2f a = *(const v2f*)(xr + kb * 4);
            if (APPLY_ACT) {
                float t0 = a.x * preScale[k0]     + preShift[k0];
                float t1 = a.y * preScale[k0 + 1] + preShift[k0 + 1];
                a.x = softplus_f(t0);
                a.y = softplus_f(t1);
            }
            // B fragment: lane holds W[k0..k0+1][n], n = nt*16 + mrow
            const float* wt = ldsWt + k0;
            v2f b0 = *(const v2f*)(wt + (mrow +  0) * FDIM);
            v2f b1 = *(const v2f*)(wt + (mrow + 16) * FDIM);
            v2f b2 = *(const v2f*)(wt + (mrow + 32) * FDIM);
            v2f b3 = *(const v2f*)(wt + (mrow + 48) * FDIM);
            acc0 = __builtin_amdgcn_wmma_f32_16x16x4_f32(false, a, false, b0, (short)0, acc0, false, false);
            acc1 = __builtin_amdgcn_wmma_f32_16x16x4_f32(false, a, false, b1, (short)0, acc1, false, false);
            acc2 = __builtin_amdgcn_wmma_f32_16x16x4_f32(false, a, false, b2, (short)0, acc2, false, false);
            acc3 = __builtin_amdgcn_wmma_f32_16x16x4_f32(false, a, false, b3, (short)0, acc3, false, false);
        }
#else
        // Scalar fallback (same accumulator layout), only if builtin absent.
        #pragma unroll
        for (int v = 0; v < 8; ++v) {
            const float* xrow = xr - kOff + (size_t)(v + halfSel * 8 - mrow) * FDIM;
            float vals[4] = {0.f, 0.f, 0.f, 0.f};
            for (int k = 0; k < FDIM; ++k) {
                float xv = xrow[k];
                if (APPLY_ACT) xv = softplus_f(xv * preScale[k] + preShift[k]);
                #pragma unroll
                for (int nt = 0; nt < 4; ++nt)
                    vals[nt] += xv * ldsWt[(nt * 16 + mrow) * FDIM + k];
            }
            acc0[v] = vals[0]; acc1[v] = vals[1]; acc2[v] = vals[2]; acc3[v] = vals[3];
        }
#endif
        // bias add + store (immediate offsets off yp) + per-feature stats
        #pragma unroll
        for (int nt = 0; nt < 4; ++nt) {
            const float bv = ldsBias[nt * 16 + mrow];
            float s = 0.f, q = 0.f;
            v8f acc = (nt == 0) ? acc0 : (nt == 1) ? acc1 : (nt == 2) ? acc2 : acc3;
            #pragma unroll
            for (int v = 0; v < 8; ++v) {
                float val = acc[v] + bv;
                yp[v * FDIM + nt * 16] = val;   // byte offset v*256 + nt*64 (immediate)
                s += val;
                q += val * val;
            }
            sumAcc[nt] += s;
            sqAcc[nt]  += q;
        }
    }

    // block-level stat reduction, then one global atomic per feature
    #pragma unroll
    for (int nt = 0; nt < 4; ++nt) {
        const int n = nt * 16 + mrow;
        atomicAdd(&ldsSum[n], sumAcc[nt]);
        atomicAdd(&ldsSq[n],  sqAcc[nt]);
    }
    __syncthreads();
    if (tid < FDIM) {
        atomicAdd(&statsOut[tid],        ldsSum[tid]);
        atomicAdd(&statsOut[FDIM + tid], ldsSq[tid]);
    }
}

// ---------------------------------------------------------------------------
// Final kernel: s = softplus(BN(y2)) . w_out + b_out ; force scatter-add
// ---------------------------------------------------------------------------
__launch_bounds__(256)
__global__ void final_force_kernel(const float* __restrict__ y2,
                                   const float* __restrict__ stats2,  // [128]
                                   const float* __restrict__ gamma2,
                                   const float* __restrict__ beta2,
                                   const float* __restrict__ w_out,
                                   const float* __restrict__ b_out,
                                   const int*   __restrict__ edge_index, // [2][E]
                                   const float* __restrict__ nbr_shift,  // [E][3]
                                   const float* __restrict__ pos,        // [N][3]
                                   float* __restrict__ force)            // [N][3]
{
    __shared__ float sScale[FDIM], sShift[FDIM], sW[FDIM];
    const int tid = threadIdx.x;
    if (tid < FDIM) {
        const float inv_e = 1.f / (float)E_EDGES;
        float mu   = stats2[tid] * inv_e;
        float var  = stats2[FDIM + tid] * inv_e - mu * mu;
        float rstd = rsqrtf(var + BN_EPS);
        float g    = gamma2[tid] * rstd;
        sScale[tid] = g;
        sShift[tid] = beta2[tid] - mu * g;
        sW[tid]     = w_out[tid];
    }
    __syncthreads();

    const float bo = b_out[0];
    const int stride = gridDim.x * blockDim.x;
    for (int e = blockIdx.x * blockDim.x + tid; e < E_EDGES; e += stride) {
        const float* yr = y2 + (size_t)e * FDIM;
        float s = 0.f;
        #pragma unroll 4
        for (int k = 0; k < FDIM; k += 4) {
            float4 v = *(const float4*)(yr + k);
            s += softplus_f(v.x * sScale[k]     + sShift[k])     * sW[k];
            s += softplus_f(v.y * sScale[k + 1] + sShift[k + 1]) * sW[k + 1];
            s += softplus_f(v.z * sScale[k + 2] + sShift[k + 2]) * sW[k + 2];
            s += softplus_f(v.w * sScale[k + 3] + sShift[k + 3]) * sW[k + 3];
        }
        s += bo;

        const int jn = edge_index[e];            // source
        const int in = edge_index[E_EDGES + e];  // target
        float dx = pos[3 * in + 0] + nbr_shift[3 * e + 0] - pos[3 * jn + 0];
        float dy = pos[3 * in + 1] + nbr_shift[3 * e + 1] - pos[3 * jn + 1];
        float dz = pos[3 * in + 2] + nbr_shift[3 * e + 2] - pos[3 * jn + 2];
        float invlen = rsqrtf(dx * dx + dy * dy + dz * dz);
        float f = s * invlen;
        atomicAdd(&force[3 * in + 0], f * dx);
        atomicAdd(&force[3 * in + 1], f * dy);
        atomicAdd(&force[3 * in + 2], f * dz);
    }
}

// ---------------------------------------------------------------------------
// Launch
// ---------------------------------------------------------------------------
extern "C" void kernel_launch(void* const* d_in, const int* in_sizes, int n_in,
                              void* d_out, int out_size, void* d_ws, size_t ws_size,
                              hipStream_t stream) {
    const float* edge_attr  = (const float*)d_in[0];
    const float* nbr_shift  = (const float*)d_in[1];
    const float* pos        = (const float*)d_in[2];
    const float* Ws         = (const float*)d_in[3]; // [3][64][64]
    const float* bs         = (const float*)d_in[4]; // [3][64]
    const float* gammas     = (const float*)d_in[5]; // [3][64]
    const float* betas      = (const float*)d_in[6]; // [3][64]
    const float* w_out      = (const float*)d_in[7]; // [64]
    const float* b_out      = (const float*)d_in[8]; // [1]
    const int*   edge_index = (const int*)d_in[9];   // [2][E]

    float* stats = (float*)d_ws;                       // 3 * 128 floats
    float* y     = (float*)((char*)d_ws + 4096);       // E * 64 floats (410 MB)
    float* force = (float*)d_out;                      // [N_NODES][3]

    // zero force + stats (buffers are poisoned between runs)
    {
        int n = out_size > 384 ? out_size : 384;
        init_zero_kernel<<<(n + 255) / 256, 256, 0, stream>>>(force, out_size, stats, 3 * 128);
    }

    const int GEMM_BLOCKS = 1024;
    // layer 0: x = edge_attr (no activation on input)
    gemm_bn_layer<0><<<GEMM_BLOCKS, 256, 0, stream>>>(
        edge_attr, y, Ws, bs, nullptr, nullptr, nullptr, stats);
    // layer 1: input = BN(stats0)+softplus of y, in place
    gemm_bn_layer<1><<<GEMM_BLOCKS, 256, 0, stream>>>(
        y, y, Ws + 64 * 64, bs + 64, gammas, betas, stats, stats + 128);
    // layer 2
    gemm_bn_layer<1><<<GEMM_BLOCKS, 256, 0, stream>>>(
        y, y, Ws + 2 * 64 * 64, bs + 128, gammas + 64, betas + 64, stats + 128, stats + 256);
    // final projection + force scatter
    final_force_kernel<<<2048, 256, 0, stream>>>(
        y, stats + 256, gammas + 128, betas + 128, w_out, b_out,
        edge_index, nbr_shift, pos, force);
}